// CANLayer_66760971649671
// MI455X (gfx1250) — compile-verified
//
#include <hip/hip_runtime.h>
#include <math.h>

#define N_CELLS 50000
#define N_EDGES 1600000
#define IN_CH   128
#define OUT_CH  32
#define HEADS   4
#define EPS_C   1.000001f   // 1.0 + 1e-6

typedef __attribute__((ext_vector_type(2))) float v2f;
typedef __attribute__((ext_vector_type(8))) float v8f;

// ---------------------------------------------------------------------------
// Kernel 1: repack weights into Wcat[128][384] = [W_low heads | W_up heads | W_skip]
// W_low/W_up are [H][IN][OUT]; Wcat[k][h*32+o] layouts them as K-major for GEMM B.
// ---------------------------------------------------------------------------
__global__ void k_repack(const float* __restrict__ Wl, const float* __restrict__ Wu,
                         const float* __restrict__ Ws, float* __restrict__ Wcat) {
    int t = blockIdx.x * blockDim.x + threadIdx.x;
    if (t >= IN_CH * 384) return;
    int k = t / 384, c = t % 384;
    float v;
    if (c < 128) {
        int h = c >> 5, o = c & 31;
        v = Wl[(h * IN_CH + k) * OUT_CH + o];
    } else if (c < 256) {
        int cc = c - 128, h = cc >> 5, o = cc & 31;
        v = Wu[(h * IN_CH + k) * OUT_CH + o];
    } else {
        v = Ws[k * 128 + (c - 256)];
    }
    Wcat[k * 384 + c] = v;
}

// ---------------------------------------------------------------------------
// Kernel 2: fused GEMM  xm/out = x[50000,128] @ Wcat[128,384] via WMMA f32 16x16x4.
// One wave computes a 16-row x 64-col strip (4 accumulators). Col strips 0..3 go
// to the xm buffer (attention features, L2-resident); strips 4..5 are the skip
// path, scaled by EPS and stored straight into d_out.
// ---------------------------------------------------------------------------
__global__ void __launch_bounds__(256) k_gemm(const float* __restrict__ x,
                                              const float* __restrict__ Wcat,
                                              float* __restrict__ xm,
                                              float* __restrict__ out) {
    const int wave = (blockIdx.x * blockDim.x + threadIdx.x) >> 5;
    const int lane = threadIdx.x & 31;
    const int NW = (N_CELLS / 16) * 6;     // 18750 tiles of 16x64
    if (wave >= NW) return;
    const int rt = wave / 6;               // row tile
    const int cs = wave % 6;               // col strip
    const int row0 = rt * 16;
    const int col0 = cs * 64;
    const int mrow = lane & 15;
    const int khalf = (lane >> 4) * 2;     // lanes 0-15 -> K {0,1}; 16-31 -> K {2,3}

    v8f acc0 = {}, acc1 = {}, acc2 = {}, acc3 = {};
    const float* xrow = x + (size_t)(row0 + mrow) * IN_CH;

    for (int k0 = 0; k0 < IN_CH; k0 += 4) {
        v2f a;
        const float2 av = *(const float2*)(xrow + k0 + khalf);
        a.x = av.x; a.y = av.y;

        const float* wr0 = Wcat + (size_t)(k0 + khalf) * 384 + col0 + mrow;
        const float* wr1 = wr0 + 384;
        v2f b0, b1, b2, b3;
        b0.x = wr0[0];  b0.y = wr1[0];
        b1.x = wr0[16]; b1.y = wr1[16];
        b2.x = wr0[32]; b2.y = wr1[32];
        b3.x = wr0[48]; b3.y = wr1[48];

        acc0 = __builtin_amdgcn_wmma_f32_16x16x4_f32(false, a, false, b0, (short)0, acc0, false, false);
        acc1 = __builtin_amdgcn_wmma_f32_16x16x4_f32(false, a, false, b1, (short)0, acc1, false, false);
        acc2 = __builtin_amdgcn_wmma_f32_16x16x4_f32(false, a, false, b2, (short)0, acc2, false, false);
        acc3 = __builtin_amdgcn_wmma_f32_16x16x4_f32(false, a, false, b3, (short)0, acc3, false, false);
    }

    // Epilogue: C/D layout — VGPR v: lanes 0-15 -> M=v, lanes 16-31 -> M=v+8; N=lane&15.
    const int rofs = (lane >> 4) * 8;
    const int nc = lane & 15;
    const bool to_skip = (col0 >= 256);   // uniform per wave
#pragma unroll
    for (int v = 0; v < 8; ++v) {
        const int row = row0 + v + rofs;
        float* xmrow = xm + (size_t)row * 256;
        float* orow  = out + (size_t)row * 128;
        float vals[4] = {acc0[v], acc1[v], acc2[v], acc3[v]};
#pragma unroll
        for (int t = 0; t < 4; ++t) {
            const int col = col0 + t * 16 + nc;
            if (to_skip) orow[col - 256] = EPS_C * vals[t];
            else         xmrow[col] = vals[t];
        }
    }
}

// ---------------------------------------------------------------------------
// Kernel 3: per-node attention score halves.  For set s (0=low,1=up), head h:
//   sSrc[s][n][h] = xm[n, s*128+h*32 .. +31] . a[h][0:32]
//   sTgt[s][n][h] = xm[...]                  . a[h][32:64]
// ---------------------------------------------------------------------------
__global__ void k_scores(const float* __restrict__ xm, const float* __restrict__ a_low,
                         const float* __restrict__ a_up, float* __restrict__ sSrc,
                         float* __restrict__ sTgt) {
    int t = blockIdx.x * blockDim.x + threadIdx.x;
    if (t >= 2 * N_CELLS) return;
    const int set = (t >= N_CELLS) ? 1 : 0;
    const int node = set ? (t - N_CELLS) : t;
    const float* a = set ? a_up : a_low;
    const float* xr = xm + (size_t)node * 256 + set * 128;
    float ssv[4], stv[4];
#pragma unroll
    for (int h = 0; h < 4; ++h) {
        const float* xh = xr + h * 32;
        const float* ah = a + h * 64;
        float s0 = 0.f, s1 = 0.f;
#pragma unroll
        for (int c = 0; c < 32; ++c) {
            const float xv = xh[c];
            s0 += xv * ah[c];
            s1 += xv * ah[32 + c];
        }
        ssv[h] = s0; stv[h] = s1;
    }
    ((float4*)sSrc)[set * N_CELLS + node] = make_float4(ssv[0], ssv[1], ssv[2], ssv[3]);
    ((float4*)sTgt)[set * N_CELLS + node] = make_float4(stv[0], stv[1], stv[2], stv[3]);
}

// ---------------------------------------------------------------------------
// Kernel 4: init segment-max (-inf) and denominator (0) buffers, both sets.
// ---------------------------------------------------------------------------
__global__ void k_init(float* __restrict__ m, float* __restrict__ den) {
    int t = blockIdx.x * blockDim.x + threadIdx.x;
    if (t >= 2 * N_CELLS * 4) return;
    m[t] = -INFINITY;
    den[t] = 0.f;
}

__device__ __forceinline__ float lrelu(float s) { return s > 0.f ? s : 0.01f * s; }

// Single-hardware-atomic float max: int-max for >=0, uint-min for <0. Valid with
// -inf init because positive floats as uint are < all negative floats as uint.
__device__ __forceinline__ void atomicMaxF(float* addr, float v) {
    if (v >= 0.f) atomicMax((int*)addr, __float_as_int(v));
    else          atomicMin((unsigned int*)addr, __float_as_uint(v));
}

// ---------------------------------------------------------------------------
// Kernel 5: segment max over edges (gridDim.y selects edge set).
// ---------------------------------------------------------------------------
__global__ void k_edge_max(const int* __restrict__ li, const float* __restrict__ lv,
                           const int* __restrict__ ui, const float* __restrict__ uv,
                           const float* __restrict__ sSrc, const float* __restrict__ sTgt,
                           float* __restrict__ m) {
    int e = blockIdx.x * blockDim.x + threadIdx.x;
    if (e >= N_EDGES) return;
    const int set = blockIdx.y;
    const int* idx = set ? ui : li;
    const float val = (set ? uv : lv)[e];
    const int i = idx[e];
    const int j = idx[N_EDGES + e];
    const float4 sj = ((const float4*)sSrc)[set * N_CELLS + j];
    const float4 si = ((const float4*)sTgt)[set * N_CELLS + i];
    const float sjv[4] = {sj.x, sj.y, sj.z, sj.w};
    const float siv[4] = {si.x, si.y, si.z, si.w};
    float* mi = m + ((size_t)set * N_CELLS + i) * 4;
#pragma unroll
    for (int h = 0; h < 4; ++h)
        atomicMaxF(mi + h, lrelu(sjv[h] + siv[h]) * val);
}

// ---------------------------------------------------------------------------
// Kernel 6: softmax denominator: den[i][h] += exp(e - m[i][h]).
// ---------------------------------------------------------------------------
__global__ void k_edge_den(const int* __restrict__ li, const float* __restrict__ lv,
                           const int* __restrict__ ui, const float* __restrict__ uv,
                           const float* __restrict__ sSrc, const float* __restrict__ sTgt,
                           const float* __restrict__ m, float* __restrict__ den) {
    int e = blockIdx.x * blockDim.x + threadIdx.x;
    if (e >= N_EDGES) return;
    const int set = blockIdx.y;
    const int* idx = set ? ui : li;
    const float val = (set ? uv : lv)[e];
    const int i = idx[e];
    const int j = idx[N_EDGES + e];
    const float4 sj = ((const float4*)sSrc)[set * N_CELLS + j];
    const float4 si = ((const float4*)sTgt)[set * N_CELLS + i];
    const float4 mi = ((const float4*)m)[set * N_CELLS + i];
    const float sjv[4] = {sj.x, sj.y, sj.z, sj.w};
    const float siv[4] = {si.x, si.y, si.z, si.w};
    const float mv[4]  = {mi.x, mi.y, mi.z, mi.w};
    float* di = den + ((size_t)set * N_CELLS + i) * 4;
#pragma unroll
    for (int h = 0; h < 4; ++h) {
        const float ev = lrelu(sjv[h] + siv[h]) * val;
        atomicAdd(di + h, __expf(ev - mv[h]));
    }
}

// ---------------------------------------------------------------------------
// Kernel 7: weighted scatter aggregation. One wave32 per edge: lane c handles
// channel c of each head; out[i][h*32+c] += alpha[h] * xm[j][set*128+h*32+c].
// xm gathers are 128B contiguous per head and L2-resident (51MB << 192MB L2).
// ---------------------------------------------------------------------------
__global__ void __launch_bounds__(256) k_edge_aggr(
        const int* __restrict__ li, const float* __restrict__ lv,
        const int* __restrict__ ui, const float* __restrict__ uv,
        const float* __restrict__ sSrc, const float* __restrict__ sTgt,
        const float* __restrict__ m, const float* __restrict__ den,
        const float* __restrict__ xm, float* __restrict__ out) {
    const int e = blockIdx.x * 8 + (threadIdx.x >> 5);
    const int lane = threadIdx.x & 31;
    if (e >= N_EDGES) return;
    const int set = blockIdx.y;
    const int* idx = set ? ui : li;
    const float val = (set ? uv : lv)[e];
    const int i = idx[e];
    const int j = idx[N_EDGES + e];
    const float4 sj = ((const float4*)sSrc)[set * N_CELLS + j];
    const float4 si = ((const float4*)sTgt)[set * N_CELLS + i];
    const float4 mi = ((const float4*)m)[set * N_CELLS + i];
    const float4 di = ((const float4*)den)[set * N_CELLS + i];
    const float sjv[4] = {sj.x, sj.y, sj.z, sj.w};
    const float siv[4] = {si.x, si.y, si.z, si.w};
    const float mv[4]  = {mi.x, mi.y, mi.z, mi.w};
    const float dv[4]  = {di.x, di.y, di.z, di.w};
    float alpha[4];
#pragma unroll
    for (int h = 0; h < 4; ++h) {
        const float ev = lrelu(sjv[h] + siv[h]) * val;
        alpha[h] = __expf(ev - mv[h]) / fmaxf(dv[h], 1e-16f);
    }
    const float* xj = xm + (size_t)j * 256 + set * 128;
    float* oi = out + (size_t)i * 128;
#pragma unroll
    for (int h = 0; h < 4; ++h)
        atomicAdd(oi + h * 32 + lane, alpha[h] * xj[h * 32 + lane]);
}

// ---------------------------------------------------------------------------
// Kernel 8: final in-place ReLU over d_out.
// ---------------------------------------------------------------------------
__global__ void k_relu(float* __restrict__ out) {
    int t = blockIdx.x * blockDim.x + threadIdx.x;
    if (t >= N_CELLS * IN_CH) return;
    out[t] = fmaxf(out[t], 0.f);
}

// ---------------------------------------------------------------------------
extern "C" void kernel_launch(void* const* d_in, const int* in_sizes, int n_in,
                              void* d_out, int out_size, void* d_ws, size_t ws_size,
                              hipStream_t stream) {
    const float* x      = (const float*)d_in[0];
    const int*   li     = (const int*)d_in[1];
    const float* lv     = (const float*)d_in[2];
    const int*   ui     = (const int*)d_in[3];
    const float* uv     = (const float*)d_in[4];
    const float* W_low  = (const float*)d_in[5];
    const float* a_low  = (const float*)d_in[6];
    const float* W_up   = (const float*)d_in[7];
    const float* a_up   = (const float*)d_in[8];
    const float* W_skip = (const float*)d_in[9];
    float* out = (float*)d_out;

    // Workspace layout (floats). Total ~57.8 MB.
    float* ws   = (float*)d_ws;
    float* Wcat = ws;                       // 128*384        = 49,152
    float* xm   = Wcat + 49152;             // 50000*256      = 12,800,000
    float* sSrc = xm + 12800000;            // 2*50000*4      = 400,000
    float* sTgt = sSrc + 400000;            // 400,000
    float* mBuf = sTgt + 400000;            // 400,000
    float* dBuf = mBuf + 400000;            // 400,000

    k_repack<<<(IN_CH * 384 + 255) / 256, 256, 0, stream>>>(W_low, W_up, W_skip, Wcat);

    const int n_waves = (N_CELLS / 16) * 6;             // 18750
    k_gemm<<<(n_waves + 7) / 8, 256, 0, stream>>>(x, Wcat, xm, out);

    k_scores<<<(2 * N_CELLS + 255) / 256, 256, 0, stream>>>(xm, a_low, a_up, sSrc, sTgt);
    k_init<<<(2 * N_CELLS * 4 + 255) / 256, 256, 0, stream>>>(mBuf, dBuf);

    dim3 eg((N_EDGES + 255) / 256, 2);
    k_edge_max<<<eg, 256, 0, stream>>>(li, lv, ui, uv, sSrc, sTgt, mBuf);
    k_edge_den<<<eg, 256, 0, stream>>>(li, lv, ui, uv, sSrc, sTgt, mBuf, dBuf);

    dim3 ag((N_EDGES + 7) / 8, 2);
    k_edge_aggr<<<ag, 256, 0, stream>>>(li, lv, ui, uv, sSrc, sTgt, mBuf, dBuf, xm, out);

    k_relu<<<(N_CELLS * IN_CH + 255) / 256, 256, 0, stream>>>(out);
}